// EquivariantCombination_60627758350459
// MI455X (gfx1250) — compile-verified
//
#include <hip/hip_runtime.h>
#include <hip/hip_bf16.h>

typedef float        v2f __attribute__((ext_vector_type(2)));
typedef float        v8f __attribute__((ext_vector_type(8)));
typedef unsigned int v4u __attribute__((ext_vector_type(4)));
typedef int          v4i __attribute__((ext_vector_type(4)));
typedef int          v8i __attribute__((ext_vector_type(8)));

#define A_DIM 1024
#define B_DIM 256
#define F_DIM 64
#define F_A_STRIDE (B_DIM * F_DIM * 3)   // 49152 floats between a-rows of f
#define F_B_STRIDE (F_DIM * 3)           // 192 floats between b-rows of f
#define T_B_STRIDE (F_DIM * 3)           // 192 floats between b-rows of tensor

#define ATILE   16
#define BSPLIT  8
#define NB      (B_DIM / BSPLIT)         // 32 b-values per block
#define CB      4                        // b-values per streamed chunk
#define NCHUNK  (NB / CB)                // 8 chunks
#define KCHUNK  (CB * 3)                 // 12 K per chunk
#define KBLOCK  (NB * 3)                 // 96 K per block

// ---- LDS map (float indices) ----
#define WK      98                       // padded K row per (F,c): 8B-aligned, conflict-free
#define WOFF    0
#define WSIZE   (F_DIM * 4 * WK)         // 25088
#define ZOFF    WSIZE                    // 2-float zero slot (8B aligned)
#define FROW    771                      // 768 data dwords + 3 TDM pad dwords
#define FB0     (WSIZE + 4)              // 25092 -> byte 100368, 16B aligned
#define FBUFSZ  (ATILE * FROW)           // 12336
#define FB1     (FB0 + FBUFSZ)
#define SMEM_FLOATS (FB1 + FBUFSZ)       // 49764 floats
#define SMEM_BYTES  (SMEM_FLOATS * 4)    // 199056 bytes (< 320KB WGP LDS)

// Issue one TDM 2D tile load: 16 a-rows x 768 floats, row stride 49152 floats,
// LDS padding: +1 dword per 256 dwords stored (interval code 7, amount code 0).
__device__ __forceinline__ void tdm_load_chunk(const float* gsrc, unsigned lds_byte_off) {
  unsigned long long ga = (unsigned long long)(uintptr_t)gsrc;
  v4u g0;
  g0.x = 1u;                                            // count=1, user descriptor
  g0.y = lds_byte_off;                                  // LDS dest (bytes)
  g0.z = (unsigned)(ga & 0xFFFFFFFFu);                  // global_addr[31:0]
  g0.w = (unsigned)((ga >> 32) & 0x01FFFFFFu) | (2u << 30); // addr[56:32] | type=2
  v8i g1;
  g1[0] = (int)((2u << 16) | (1u << 20) | (7u << 22));  // data_size=4B, pad_en, interval=256dw, amount=1dw
  g1[1] = (int)(((unsigned)(CB * F_DIM * 3) & 0xFFFFu) << 16); // tensor_dim0 lo16 (768)
  g1[2] = (int)(((unsigned)ATILE & 0xFFFFu) << 16);     // tensor_dim0 hi=0 | tensor_dim1 lo16 (16)
  g1[3] = (int)(((unsigned)(CB * F_DIM * 3)) << 16);    // tensor_dim1 hi=0 | tile_dim0 (768)
  g1[4] = ATILE;                                        // tile_dim1=16, tile_dim2=0
  g1[5] = F_A_STRIDE;                                   // tensor_dim0_stride lo32 (49152)
  g1[6] = 0;                                            // stride hi | dim1_stride lo
  g1[7] = 0;
  v4i gz4  = {0, 0, 0, 0};
  v8i gz8  = {0, 0, 0, 0, 0, 0, 0, 0};
  // 6-arg form on this toolchain: (g0, g1, g2, g3, g4, cpol)
  __builtin_amdgcn_tensor_load_to_lds(g0, g1, gz4, gz4, gz8, 0);
}

__global__ void __launch_bounds__(256, 1)
eqcomb_kernel(const float* __restrict__ tensor, const float* __restrict__ f,
              float* __restrict__ out) {
  extern __shared__ float smem[];
  const int tid  = threadIdx.x;
  const int lane = tid & 31;
  const int wid  = tid >> 5;
  const int bid  = blockIdx.x;
  const int a0   = (bid & 63) * ATILE;
  const int b0   = (bid >> 6) * NB;

  // ---- precompute W[F][c][K] for this block's 96 K values ----
  for (int idx = tid; idx < F_DIM * KBLOCK; idx += 256) {
    const int Fv = idx / KBLOCK;
    const int K  = idx - Fv * KBLOCK;
    const int b  = K / 3;
    const int j  = K - 3 * b;
    const float* t = tensor + (size_t)(b0 + b) * T_B_STRIDE + Fv * 3;
    const float t0 = t[0], t1 = t[1], t2 = t[2];
    float* wr = smem + WOFF + (Fv * 4) * WK + K;
    wr[0 * WK] = t0 + t1 + t2;                               // o0 coefficient
    wr[1 * WK] = (j == 1) ? t2 : ((j == 2) ? -t1 : 0.f);     // (f x t).x
    wr[2 * WK] = (j == 2) ? t0 : ((j == 0) ? -t2 : 0.f);     // (f x t).y
    wr[3 * WK] = (j == 0) ? t1 : ((j == 1) ? -t0 : 0.f);     // (f x t).z
  }
  if (tid == 0) { smem[ZOFF] = 0.f; smem[ZOFF + 1] = 0.f; }

  // lane geometry for V_WMMA_F32_16X16X4_F32
  const int aL = lane & 15;         // A-matrix row (a index)
  const int kk = (lane >> 4) * 2;   // lanes 0-15 -> K {0,1}; lanes 16-31 -> K {2,3}
  const int cN = lane & 15;         // B/D column (output channel, only 0..3 live)

  const v8f vzero = {0.f, 0.f, 0.f, 0.f, 0.f, 0.f, 0.f, 0.f};
  v8f acc[8];
#pragma unroll
  for (int i = 0; i < 8; ++i) acc[i] = vzero;

  // prime the pipeline: chunk 0 into buffer 0
  if (wid == 0) {
    tdm_load_chunk(f + (size_t)a0 * F_A_STRIDE + (size_t)b0 * F_B_STRIDE,
                   (unsigned)(uintptr_t)(smem + FB0));
  }

  for (int ch = 0; ch < NCHUNK; ++ch) {
    if (wid == 0) {
      if (ch + 1 < NCHUNK) {
        tdm_load_chunk(
            f + (size_t)a0 * F_A_STRIDE + (size_t)(b0 + (ch + 1) * CB) * F_B_STRIDE,
            (unsigned)(uintptr_t)(smem + (((ch + 1) & 1) ? FB1 : FB0)));
        __builtin_amdgcn_s_wait_tensorcnt(1);   // chunk ch complete (in-order)
      } else {
        __builtin_amdgcn_s_wait_tensorcnt(0);
      }
    }
    __syncthreads();   // chunk ch visible (and W on first pass)

    const float* fb = smem + ((ch & 1) ? FB1 : FB0);

#pragma unroll
    for (int fi = 0; fi < 8; ++fi) {
      const int Fv = wid * 8 + fi;
      const float* wbase = (cN < 4) ? (smem + WOFF + (size_t)(Fv * 4 + cN) * WK)
                                    : (smem + ZOFF);
      const int wstep = (cN < 4) ? 1 : 0;   // zero-slot lanes stay on the zero slot
#pragma unroll
      for (int st = 0; st < 3; ++st) {
        const int Kl0 = st * 4 + kk;        // chunk-local K for this lane
        const int Kl1 = Kl0 + 1;
        const int bl0 = Kl0 / 3, j0 = Kl0 - 3 * bl0;
        const int bl1 = Kl1 / 3, j1 = Kl1 - 3 * bl1;
        int off0 = bl0 * F_B_STRIDE + Fv * 3 + j0; off0 += (off0 >> 8); // TDM pad skew
        int off1 = bl1 * F_B_STRIDE + Fv * 3 + j1; off1 += (off1 >> 8);
        v2f av;
        av.x = fb[aL * FROW + off0];
        av.y = fb[aL * FROW + off1];
        const int Kg = ch * KCHUNK + st * 4 + kk;      // even -> 8B-aligned v2f read
        const v2f wv = *(const v2f*)(wbase + wstep * Kg);
        acc[fi] = __builtin_amdgcn_wmma_f32_16x16x4_f32(
            false, av, false, wv, (short)0, acc[fi], false, false);
      }
    }
    __syncthreads();   // finish reading buf before chunk ch+2 overwrites it
  }

  // ---- writeout: D VGPR r holds (a = r + 8*(lane>=16), c = lane&15) ----
#pragma unroll
  for (int fi = 0; fi < 8; ++fi) {
    const int Fv = wid * 8 + fi;
    if (cN < 4) {
#pragma unroll
      for (int r = 0; r < 8; ++r) {
        const int ag = a0 + r + ((lane >> 4) << 3);
        const float v = acc[fi][r];
        float* dst = (cN == 0)
            ? (out + (size_t)ag * F_DIM + Fv)                                 // o0 [A,F,1]
            : (out + (size_t)A_DIM * F_DIM + ((size_t)ag * F_DIM + Fv) * 3 + (cN - 1)); // o1
        atomicAdd(dst, v);
      }
    }
  }
}

__global__ void zero_out_kernel(float* __restrict__ out, int n) {
  int i = blockIdx.x * blockDim.x + threadIdx.x;
  const int stride = gridDim.x * blockDim.x;
  for (; i < n; i += stride) out[i] = 0.f;
}

extern "C" void kernel_launch(void* const* d_in, const int* in_sizes, int n_in,
                              void* d_out, int out_size, void* d_ws, size_t ws_size,
                              hipStream_t stream) {
  (void)in_sizes; (void)n_in; (void)d_ws; (void)ws_size;
  const float* tensor = (const float*)d_in[0];   // [256, 64, 3]
  const float* f      = (const float*)d_in[1];   // [1024, 256, 64, 3]
  float* out = (float*)d_out;                    // o0 [1024,64,1] ++ o1 [1024,64,3]

  // allow >64KB dynamic LDS (no-op under compile-only, harmless during capture)
  (void)hipFuncSetAttribute((const void*)eqcomb_kernel,
                            hipFuncAttributeMaxDynamicSharedMemorySize, SMEM_BYTES);

  zero_out_kernel<<<256, 256, 0, stream>>>(out, out_size);
  eqcomb_kernel<<<dim3(64 * BSPLIT), dim3(256), SMEM_BYTES, stream>>>(tensor, f, out);
}